// GraphLaplacianLoss_73796128080164
// MI455X (gfx1250) — compile-verified
//
#include <hip/hip_runtime.h>
#include <hip/hip_bf16.h>
#include <stdint.h>

#define B_ 64
#define V_ 100000
#define F_ 200000
#define TILE_F 1024
#define THREADS 256

typedef float v8f  __attribute__((ext_vector_type(8)));
typedef float v2f  __attribute__((ext_vector_type(2)));
typedef unsigned int u32x4 __attribute__((ext_vector_type(4)));
typedef int i32x8 __attribute__((ext_vector_type(8)));
typedef int i32x4 __attribute__((ext_vector_type(4)));

// Fire-and-forget f32 atomic add (no return -> STOREcnt only, max issue rate).
// Device scope so the L2 atomic units keep cross-WGP coherence.
__device__ __forceinline__ void gAtomicAddF32(float* p, float v) {
    asm volatile("global_atomic_add_f32 %0, %1, off scope:SCOPE_DEV"
                 :: "v"(p), "v"(v) : "memory");
}

__global__ __launch_bounds__(THREADS) void zero_kernel(float* p, long long n) {
    long long i = (long long)blockIdx.x * blockDim.x + threadIdx.x;
    long long stride = (long long)gridDim.x * blockDim.x;
    for (; i < n; i += stride) p[i] = 0.f;
}

// deg[v] += 2 for each face containing v (each face vertex is a src twice).
__global__ __launch_bounds__(THREADS) void deg_kernel(const int* __restrict__ faces,
                                                      float* __restrict__ deg) {
    int f = blockIdx.x * blockDim.x + threadIdx.x;
    if (f >= F_) return;
    int i = faces[3 * f + 0];
    int j = faces[3 * f + 1];
    int k = faces[3 * f + 2];
    gAtomicAddF32(deg + i, 2.f);
    gAtomicAddF32(deg + j, 2.f);
    gAtomicAddF32(deg + k, 2.f);
}

// One block = (face tile, batch). Face tile staged to LDS once via TDM,
// then 9 pair-sum atomics per face into nbr_sum (L2-resident).
__global__ __launch_bounds__(THREADS) void scatter_kernel(const float* __restrict__ verts,
                                                          const int* __restrict__ faces,
                                                          float* __restrict__ nbr,
                                                          int b0) {
    __shared__ int sF[TILE_F * 3];

    const int t0 = blockIdx.x * TILE_F;
    const int nF = min(TILE_F, F_ - t0);
    const int b  = b0 + blockIdx.y;

    // ---- TDM async stage of nF*3 dwords of face indices into LDS ----
    if (threadIdx.x < 32) {
        uint64_t ga  = (uint64_t)(const void*)(faces + (size_t)t0 * 3);
        uint32_t lds = (uint32_t)(uintptr_t)(void*)sF;   // flat-addr low 32b == LDS byte offset
        uint32_t ndw = (uint32_t)nF * 3u;

        u32x4 g0;
        g0[0] = (unsigned)__builtin_amdgcn_readfirstlane((int)1u);                 // count=1
        g0[1] = (unsigned)__builtin_amdgcn_readfirstlane((int)lds);                // lds_addr
        g0[2] = (unsigned)__builtin_amdgcn_readfirstlane((int)(uint32_t)ga);       // gaddr[31:0]
        g0[3] = (unsigned)__builtin_amdgcn_readfirstlane(
                    (int)(((uint32_t)(ga >> 32) & 0x01FFFFFFu) | 0x80000000u));    // gaddr[56:32] | type=2

        i32x8 g1;
        g1[0] = __builtin_amdgcn_readfirstlane((int)(2u << 16));                   // data_size=4B, mask=0
        g1[1] = __builtin_amdgcn_readfirstlane((int)((ndw & 0xFFFFu) << 16));      // tensor_dim0 lo
        g1[2] = __builtin_amdgcn_readfirstlane((int)(ndw >> 16));                  // tensor_dim0 hi, dim1=0
        g1[3] = __builtin_amdgcn_readfirstlane((int)((ndw & 0xFFFFu) << 16));      // tile_dim0 = ndw
        g1[4] = 0;                                                                 // tile_dim1/2 unused
        g1[5] = __builtin_amdgcn_readfirstlane((int)ndw);                          // dim0 stride lo
        g1[6] = 0;
        g1[7] = 0;
        i32x4 gz4 = {0, 0, 0, 0};
        i32x8 gz8 = {0, 0, 0, 0, 0, 0, 0, 0};
        __builtin_amdgcn_tensor_load_to_lds(g0, g1, gz4, gz4, gz8, 0);
        __builtin_amdgcn_s_wait_tensorcnt(0);
    }
    __syncthreads();

    const float* __restrict__ vb = verts + (size_t)b * (V_ * 3);
    float* __restrict__ nb = nbr + (size_t)blockIdx.y * (V_ * 3);

    for (int f = threadIdx.x; f < nF; f += THREADS) {
        int i = sF[3 * f + 0];
        int j = sF[3 * f + 1];
        int k = sF[3 * f + 2];
        float ix = vb[3 * i + 0], iy = vb[3 * i + 1], iz = vb[3 * i + 2];
        float jx = vb[3 * j + 0], jy = vb[3 * j + 1], jz = vb[3 * j + 2];
        float kx = vb[3 * k + 0], ky = vb[3 * k + 1], kz = vb[3 * k + 2];
        // nbr[i] += v[j]+v[k]; nbr[j] += v[i]+v[k]; nbr[k] += v[i]+v[j]
        gAtomicAddF32(nb + 3 * i + 0, jx + kx);
        gAtomicAddF32(nb + 3 * i + 1, jy + ky);
        gAtomicAddF32(nb + 3 * i + 2, jz + kz);
        gAtomicAddF32(nb + 3 * j + 0, ix + kx);
        gAtomicAddF32(nb + 3 * j + 1, iy + ky);
        gAtomicAddF32(nb + 3 * j + 2, iz + kz);
        gAtomicAddF32(nb + 3 * k + 0, ix + jx);
        gAtomicAddF32(nb + 3 * k + 1, iy + jy);
        gAtomicAddF32(nb + 3 * k + 2, iz + jz);
    }
}

// per_vert = ||v - nbr/max(deg,1)||; mean via WMMA wave reduction + block atomic.
__global__ __launch_bounds__(THREADS) void finalize_kernel(const float* __restrict__ verts,
                                                           const float* __restrict__ nbr,
                                                           const float* __restrict__ deg,
                                                           float* __restrict__ out,
                                                           int b0, int cur) {
    const int total = cur * V_;
    float s = 0.f;
    for (int idx = blockIdx.x * blockDim.x + threadIdx.x; idx < total;
         idx += gridDim.x * blockDim.x) {
        int bc = idx / V_;
        int v  = idx - bc * V_;
        const float* pv = verts + ((size_t)(b0 + bc) * V_ + v) * 3;
        const float* pn = nbr + ((size_t)bc * V_ + v) * 3;
        float d = deg[v];
        d = d > 1.f ? d : 1.f;
        float inv = 1.f / d;
        float lx = pv[0] - pn[0] * inv;
        float ly = pv[1] - pn[1] * inv;
        float lz = pv[2] - pn[2] * inv;
        s += sqrtf(lx * lx + ly * ly + lz * lz);
    }

    // Wave32 reduction with the matrix engine: A lane-striped partials (K=0 slot),
    // B = ones  =>  D[m,*] = s[m] + s[m+16]  (fp32-exact, one v_wmma op).
    v2f a;    a[0] = s;   a[1] = 0.f;
    v2f bone; bone[0] = 1.f; bone[1] = 1.f;
    v8f c = {};
    v8f dm = __builtin_amdgcn_wmma_f32_16x16x4_f32(false, a, false, bone,
                                                   (short)0, c, false, false);
    float p = dm[0] + dm[1] + dm[2] + dm[3] + dm[4] + dm[5] + dm[6] + dm[7];
    float waveSum = __shfl(p, 0, 32) + __shfl(p, 16, 32);

    __shared__ float sW[THREADS / 32];
    int lane = threadIdx.x & 31;
    int wid  = threadIdx.x >> 5;
    if (lane == 0) sW[wid] = waveSum;
    __syncthreads();
    if (threadIdx.x == 0) {
        float bs = 0.f;
#pragma unroll
        for (int w = 0; w < THREADS / 32; ++w) bs += sW[w];
        gAtomicAddF32(out, bs * (1.0f / ((float)B_ * (float)V_)));
    }
}

static inline int imin_(int a, int b) { return a < b ? a : b; }

extern "C" void kernel_launch(void* const* d_in, const int* in_sizes, int n_in,
                              void* d_out, int out_size, void* d_ws, size_t ws_size,
                              hipStream_t stream) {
    const float* verts = (const float*)d_in[0];
    const int*   faces = (const int*)d_in[1];
    float* out = (float*)d_out;

    float* deg = (float*)d_ws;
    size_t degB = ((size_t)V_ * 4 + 255) & ~(size_t)255;
    float* nbr = (float*)((char*)d_ws + degB);
    size_t rem = ws_size > degB ? ws_size - degB : 0;
    size_t sliceB = (size_t)V_ * 3 * 4;
    int bc = (int)(rem / sliceB);
    if (bc > B_) bc = B_;
    if (bc < 1) bc = 1;

    zero_kernel<<<512, THREADS, 0, stream>>>(deg, (long long)V_);
    zero_kernel<<<1, 32, 0, stream>>>(out, 1LL);
    deg_kernel<<<(F_ + THREADS - 1) / THREADS, THREADS, 0, stream>>>(faces, deg);

    const int faceTiles = (F_ + TILE_F - 1) / TILE_F;
    for (int b0 = 0; b0 < B_; b0 += bc) {
        int cur = imin_(bc, B_ - b0);
        zero_kernel<<<2048, THREADS, 0, stream>>>(nbr, (long long)cur * V_ * 3);
        dim3 g(faceTiles, cur);
        scatter_kernel<<<g, THREADS, 0, stream>>>(verts, faces, nbr, b0);
        finalize_kernel<<<2048, THREADS, 0, stream>>>(verts, nbr, deg, out, b0, cur);
    }
}